// QSplitShiftmax_26173530702399
// MI455X (gfx1250) — compile-verified
//
#include <hip/hip_runtime.h>
#include <stdint.h>

// QSplitShiftmax: row-wise integer softmax approximation with A/B split
// requantization. Memory-bound streaming op (512 MB traffic, ~22us floor at
// 23.3 TB/s HBM). One wave32 per 1024-float row; CDNA5 async global->LDS
// staging (global_load_async_to_lds_b128 + s_wait_asynccnt) with NT temporal
// hints, wave shuffle reductions, nontemporal 128-bit stores. No matmul in
// the op, so WMMA is intentionally not used (pure overhead for this roofline).

#define N_SHIFT   20
#define ROW_LEN   1024
#define ROWS_PER_BLOCK 8
#define TWO32F    4294967296.0f

typedef __attribute__((ext_vector_type(4))) float v4f;

__global__ __launch_bounds__(256) void qsplit_shiftmax_kernel(
    const float* __restrict__ x,
    const float* __restrict__ scale_p,
    const float* __restrict__ thr_p,
    float* __restrict__ out,
    int nrows)
{
  __shared__ __align__(16) float lds[ROWS_PER_BLOCK * ROW_LEN];  // 32 KB

  const int wave = threadIdx.x >> 5;   // 0..7
  const int lane = threadIdx.x & 31;   // wave32
  const int row  = blockIdx.x * ROWS_PER_BLOCK + wave;
  if (row >= nrows) return;            // whole-wave uniform exit

  const float scale = scale_p[0];
  const float thr   = thr_p[0];

  const float* __restrict__ rowp = x   + (size_t)row * ROW_LEN;
  float*       __restrict__ outp = out + (size_t)row * ROW_LEN;

  // ---- Async global -> LDS staging (CDNA5 async data mover) -------------
  unsigned lds_base;
  {
    __attribute__((address_space(3))) float* lp =
        (__attribute__((address_space(3))) float*)(&lds[wave * ROW_LEN]);
    lds_base = (unsigned)(uintptr_t)lp;
  }
  const unsigned vdst  = lds_base + (unsigned)lane * 16u;           // LDS addr/lane
  const uint64_t gaddr = (uint64_t)(uintptr_t)rowp + (uint64_t)lane * 16u;

  // 8 x b128 per lane covers the 4 KB row; the immediate offset advances BOTH
  // the global and the LDS address.
  // NT hint: data is streamed exactly once, don't let it camp in cache.
  asm volatile(
      "global_load_async_to_lds_b128 %0, %1, off offset:0 th:TH_LOAD_NT\n\t"
      "global_load_async_to_lds_b128 %0, %1, off offset:512 th:TH_LOAD_NT\n\t"
      "global_load_async_to_lds_b128 %0, %1, off offset:1024 th:TH_LOAD_NT\n\t"
      "global_load_async_to_lds_b128 %0, %1, off offset:1536 th:TH_LOAD_NT\n\t"
      "global_load_async_to_lds_b128 %0, %1, off offset:2048 th:TH_LOAD_NT\n\t"
      "global_load_async_to_lds_b128 %0, %1, off offset:2560 th:TH_LOAD_NT\n\t"
      "global_load_async_to_lds_b128 %0, %1, off offset:3072 th:TH_LOAD_NT\n\t"
      "global_load_async_to_lds_b128 %0, %1, off offset:3584 th:TH_LOAD_NT\n\t"
      "s_wait_asynccnt 0"
      :
      : "v"(vdst), "v"(gaddr)
      : "memory");

  // ---- Pass 1: read row from LDS, scale, row max (4-way tree) -----------
  const v4f* __restrict__ lrow = (const v4f*)(lds + wave * ROW_LEN);
  const float inv_s = 1.0f / scale;

  float xv[32];
  float m0 = -3.402823466e+38f, m1 = m0, m2 = m0, m3 = m0;
#pragma unroll
  for (int j = 0; j < 8; ++j) {
    v4f t = lrow[j * 32 + lane];
    float a0 = t.x * inv_s, a1 = t.y * inv_s;
    float a2 = t.z * inv_s, a3 = t.w * inv_s;
    xv[4 * j + 0] = a0; xv[4 * j + 1] = a1;
    xv[4 * j + 2] = a2; xv[4 * j + 3] = a3;
    m0 = fmaxf(m0, a0); m1 = fmaxf(m1, a1);
    m2 = fmaxf(m2, a2); m3 = fmaxf(m3, a3);
  }
  float m = fmaxf(fmaxf(m0, m1), fmaxf(m2, m3));
#pragma unroll
  for (int off = 16; off > 0; off >>= 1)
    m = fmaxf(m, __shfl_xor(m, off, 32));

  // ---- Pass 2: shift-exp per element + row sum (4 accumulators) ---------
  const float x0     = floorf(-1.0f / scale);        // integer -1 in scale
  const float rcp_x0 = 1.0f / x0;
  const float lo     = (float)N_SHIFT * x0;

  float s0 = 0.0f, s1 = 0.0f, s2 = 0.0f, s3 = 0.0f;
#pragma unroll
  for (int i = 0; i < 32; ++i) {
    float xi = xv[i] - m;
    xi = xi + floorf(xi * 0.5f) - floorf(xi * 0.0625f);  // x + x/2 - x/16
    xi = fmaxf(xi, lo);
    float q = floorf(xi * rcp_x0);                       // q in [0, 20]
    float r = xi - x0 * q;                               // x0*q exact
    float e = r * 0.5f - x0;
    e = floorf(__builtin_ldexpf(e, N_SHIFT - (int)q));   // v_ldexp_f32
    e = fmaxf(e, 0.0f);
    xv[i] = e;
    if      ((i & 3) == 0) s0 += e;
    else if ((i & 3) == 1) s1 += e;
    else if ((i & 3) == 2) s2 += e;
    else                   s3 += e;
  }
  float s = (s0 + s1) + (s2 + s3);
#pragma unroll
  for (int off = 16; off > 0; off >>= 1)
    s += __shfl_xor(s, off, 32);

  // ---- Pass 3: A/B split requantization, nontemporal store --------------
  s = fminf(s, TWO32F);
  const float factor = floorf(TWO32F / s);
  const float osA = thr / 255.0f;
  const float osB = (1.0f - thr) / 255.0f;
  const float at  = floorf(thr * 256.0f) * s * (1.0f / 256.0f);  // /256 exact
  const float rA  = 1.0f / (TWO32F * osA);
  const float rB  = 1.0f / (TWO32F * osB);

  v4f* __restrict__ orow = (v4f*)outp;
#pragma unroll
  for (int j = 0; j < 8; ++j) {
    v4f o;
#pragma unroll
    for (int k = 0; k < 4; ++k) {
      float e  = xv[4 * j + k];
      float ef = e * factor;
      float a  = floorf(ef * rA) * osA;
      float b  = fminf(floorf(ef * rB), 255.0f) * osB;
      o[k] = (e <= at) ? a : b;
    }
    __builtin_nontemporal_store(o, &orow[j * 32 + lane]);
  }
}

extern "C" void kernel_launch(void* const* d_in, const int* in_sizes, int n_in,
                              void* d_out, int out_size, void* d_ws, size_t ws_size,
                              hipStream_t stream) {
  (void)n_in; (void)out_size; (void)d_ws; (void)ws_size;
  const float* x     = (const float*)d_in[0];
  const float* scale = (const float*)d_in[1];
  const float* thr   = (const float*)d_in[2];
  float* out         = (float*)d_out;

  const int total = in_sizes[0];           // 4*16*1024*1024
  const int nrows = total / ROW_LEN;       // 65536 rows of 1024
  const int blocks = (nrows + ROWS_PER_BLOCK - 1) / ROWS_PER_BLOCK;

  qsplit_shiftmax_kernel<<<blocks, 256, 0, stream>>>(x, scale, thr, out, nrows);
}